// deepsignet_5231270167236
// MI455X (gfx1250) — compile-verified
//
#include <hip/hip_runtime.h>
#include <math.h>

// ---------------------------------------------------------------------------
// DeepSigNet for MI455X (gfx1250, wave32).
//   Phase 1: depth-5 path signature via Chen's relation, one workgroup per
//            batch element. Levels 1-4 live in LDS; level 5 (3125 elems) is
//            REGISTER-RESIDENT (13 VGPRs/lane), since s5[q] is touched only
//            by its owning thread. Levels updated 5->1 in place (level k
//            reads only old levels < k). Output rows padded to 3908 floats.
//   Phase 2: w0 transposed+padded to (32 x 3908) in ws, then
//            sig(256x3908) @ w0 -> h0 with V_WMMA_F32_16X16X4_F32;
//            branch-free K loop, all fragment loads aligned global b64.
//   Phase 3: 4x (32x32) ReLU layers (weights transposed into LDS, all
//            fragment traffic ds b64) + 32->1 sigmoid head.
// Workspace (floats): sig[256*3908] | w0t[32*3908] | h0[256*32]  (~4.5 MB)
// ---------------------------------------------------------------------------

#define BATCH 256
#define LEN   512
#define NSTEP 511          // LEN-1 increments
#define SIGDIM 3905        // 5+25+125+625+3125
#define SIGP   3908        // padded K (multiple of 4 floats = 16B)
#define NQ5    13          // ceil(3125/256) level-5 elements per thread

typedef __attribute__((ext_vector_type(2))) float v2f;
typedef __attribute__((ext_vector_type(8))) float v8f;

// ---------------------------------------------------------------------------
// Kernel 1: conv features + depth-5 signature scan. grid=BATCH, block=256.
// ---------------------------------------------------------------------------
__global__ __launch_bounds__(256) void sig5_kernel(
    const float* __restrict__ x,        // (B,1,L)
    const float* __restrict__ conv_w,   // (3,1,3)
    const float* __restrict__ conv_b,   // (3,)
    float* __restrict__ sig_out)        // (B, SIGP), zero padded tail
{
    __shared__ float xs[LEN];
    __shared__ float inc[NSTEP * 5];    // increments, row-major (t, channel)
    __shared__ float sig[780];          // s1|s2|s3|s4  (level 5 in registers)
    __shared__ float e2[25];
    __shared__ float e3[125];
    __shared__ float e4[625];

    const int tid = threadIdx.x;
    const int b   = blockIdx.x;
    const float* xrow = x + b * LEN;

    for (int i = tid; i < LEN; i += 256) xs[i] = xrow[i];
    for (int i = tid; i < 780; i += 256) sig[i] = 0.0f;

    float r5[NQ5];                       // this thread's s5 elements
#pragma unroll
    for (int j = 0; j < NQ5; ++j) r5[j] = 0.0f;
    __syncthreads();

    float w00 = conv_w[0], w01 = conv_w[1], w02 = conv_w[2];
    float w10 = conv_w[3], w11 = conv_w[4], w12 = conv_w[5];
    float w20 = conv_w[6], w21 = conv_w[7], w22 = conv_w[8];

    // increments: conv channels (bias cancels), raw x, constant time channel
    for (int t = tid; t < NSTEP; t += 256) {
        float xm1 = (t >= 1)      ? xs[t - 1] : 0.0f;
        float x0  = xs[t];
        float x1  = xs[t + 1];
        float x2  = (t + 2 < LEN) ? xs[t + 2] : 0.0f;
        inc[t * 5 + 0] = (w00 * x0 + w01 * x1 + w02 * x2) - (w00 * xm1 + w01 * x0 + w02 * x1);
        inc[t * 5 + 1] = (w10 * x0 + w11 * x1 + w12 * x2) - (w10 * xm1 + w11 * x0 + w12 * x1);
        inc[t * 5 + 2] = (w20 * x0 + w21 * x1 + w22 * x2) - (w20 * xm1 + w21 * x0 + w22 * x1);
        inc[t * 5 + 3] = x1 - x0;
        inc[t * 5 + 4] = 1.0f / (float)(LEN - 1);
    }
    __syncthreads();

    float* s1 = sig;
    float* s2 = sig + 5;
    float* s3 = sig + 30;
    float* s4 = sig + 155;

    for (int t = 0; t < NSTEP; ++t) {
        const float* v = &inc[t * 5];

        // exponentials of the segment: e_k = e_{k-1} (x) v / k
        if (tid < 25)  e2[tid] = v[tid / 5] * v[tid % 5] * 0.5f;
        __syncthreads();
        if (tid < 125) e3[tid] = e2[tid / 5] * v[tid % 5] * (1.0f / 3.0f);
        __syncthreads();
        for (int q = tid; q < 625; q += 256) e4[q] = e3[q / 5] * v[q % 5] * 0.25f;
        __syncthreads();

        // level 5 (registers): n5 = s5 + s4(x)e1 + s3(x)e2 + s2(x)e3
        //                           + s1(x)e4 + e5,   e5 = e4 (x) v / 5
#pragma unroll
        for (int j = 0; j < NQ5; ++j) {
            int q = tid + 256 * j;
            if (q < 3125) {
                int a5 = q / 5,   b5 = q - a5 * 5;
                int a25 = q / 25,  b25 = q - a25 * 25;
                int a125 = q / 125, b125 = q - a125 * 125;
                int a625 = q / 625, b625 = q - a625 * 625;
                r5[j] += (s4[a5] + 0.2f * e4[a5]) * v[b5]
                       + s3[a25]  * e2[b25]
                       + s2[a125] * e3[b125]
                       + s1[a625] * e4[b625];
            }
        }
        __syncthreads();   // level-5 reads of s4..s1 done before s4 is written

        // level 4
        for (int q = tid; q < 625; q += 256) {
            int a5 = q / 5,  b5 = q - a5 * 5;
            int a25 = q / 25, b25 = q - a25 * 25;
            int a125 = q / 125, b125 = q - a125 * 125;
            s4[q] += (s3[a5] + 0.25f * e3[a5]) * v[b5]
                   + s2[a25]  * e2[b25]
                   + s1[a125] * e3[b125];
        }
        __syncthreads();

        // level 3
        if (tid < 125) {
            int a5 = tid / 5, b5 = tid - a5 * 5;
            int a25 = tid / 25, b25 = tid - a25 * 25;
            s3[tid] += (s2[a5] + (1.0f / 3.0f) * e2[a5]) * v[b5]
                     + s1[a25] * e2[b25];
        }
        __syncthreads();

        // levels 2 & 1 entirely in wave 0: wave program order keeps the s1
        // read (in the s2 update) before the s1 write.
        if (tid < 25) s2[tid] += s1[tid / 5] * v[tid % 5] + v[tid / 5] * v[tid % 5] * 0.5f;
        if (tid < 5)  s1[tid] += v[tid];
        __syncthreads();
    }

    // write out: s1..s4 from LDS, s5 from registers, zero pad tail
    float* out = sig_out + b * SIGP;
    for (int q = tid; q < 780; q += 256) out[q] = sig[q];
#pragma unroll
    for (int j = 0; j < NQ5; ++j) {
        int q = tid + 256 * j;
        if (q < 3125) out[780 + q] = r5[j];
    }
    if (tid < SIGP - SIGDIM) out[SIGDIM + tid] = 0.0f;
}

// ---------------------------------------------------------------------------
// Kernel 1b: transpose + zero-pad w0 (3905x32) -> w0t (32 x SIGP).
// ---------------------------------------------------------------------------
__global__ __launch_bounds__(256) void pack_w0t_kernel(
    const float* __restrict__ w0, float* __restrict__ w0t)
{
    int i = blockIdx.x * 256 + threadIdx.x;       // over 32*SIGP
    if (i < 32 * SIGP) {
        int n = i / SIGP;
        int k = i - n * SIGP;
        w0t[i] = (k < SIGDIM) ? w0[k * 32 + n] : 0.0f;
    }
}

// ---------------------------------------------------------------------------
// WMMA helper: one 16x16x4 f32 step.
// A frag (16x4): lanes 0-15 -> M=lane, K={0,1}; lanes 16-31 -> K={2,3}.
// B frag (4x16): lanes 0-15 -> N=lane, K={0,1}; lanes 16-31 -> K={2,3}.
// D (16x16 f32): 8 VGPRs, M = r (+8 upper half-wave), N = lane&15.
// ---------------------------------------------------------------------------
__device__ inline v8f wmma_f32_k4(v2f a, v2f bm, v8f c) {
    return __builtin_amdgcn_wmma_f32_16x16x4_f32(
        /*neg_a=*/false, a, /*neg_b=*/false, bm,
        /*c_mod=*/(short)0, c, /*reuse_a=*/false, /*reuse_b=*/false);
}

// ---------------------------------------------------------------------------
// Kernel 2: h0 = relu(sig @ w0 + b0).  256 x SIGP x 32, f32 WMMA.
// grid=4, block=256 (8 waves) -> 32 waves, one 16x16 output tile per wave.
// Branch-free K loop: 977 iterations of {2x global_load_b64, 1x wmma}.
// ---------------------------------------------------------------------------
__global__ __launch_bounds__(256) void gemm0_kernel(
    const float* __restrict__ sig,   // (256, SIGP) zero-padded
    const float* __restrict__ w0t,   // (32, SIGP)  transposed, zero-padded
    const float* __restrict__ bias0, // (32,)
    float* __restrict__ h0)          // (256, 32)
{
    const int lane = threadIdx.x & 31;
    const int wv   = threadIdx.x >> 5;
    const int tile = blockIdx.x * 8 + wv;   // 0..31
    const int tm   = tile & 15;             // 16 row tiles
    const int tn   = tile >> 4;             // 2  col tiles
    const int rm   = tm * 16;
    const int cn   = tn * 16;
    const int half = lane >> 4;             // which K pair this lane holds
    const int lr   = lane & 15;

    const int m = rm + lr;                  // A row for this lane
    const int n = cn + lr;                  // B col for this lane

    // float2 views; rows are 16B-aligned (SIGP even), K offsets even.
    const float2* Arow = (const float2*)(sig + (size_t)m * SIGP) + half;
    const float2* Brow = (const float2*)(w0t + (size_t)n * SIGP) + half;

    v8f acc = {};
#pragma unroll 4
    for (int k2 = 0; k2 < SIGP / 2; k2 += 2) {   // k2 in float2 units, 4 floats/step
        float2 a  = Arow[k2];
        float2 bq = Brow[k2];
        v2f A  = {a.x, a.y};
        v2f Bf = {bq.x, bq.y};
        acc = wmma_f32_k4(A, Bf, acc);
    }

    float bn = bias0[n];
#pragma unroll
    for (int r = 0; r < 8; ++r) {
        int mo = rm + r + half * 8;
        float vv = acc[r] + bn;
        h0[mo * 32 + n] = vv > 0.0f ? vv : 0.0f;
    }
}

// ---------------------------------------------------------------------------
// Kernel 3: four 32x32 ReLU layers (weights transposed into LDS, WMMA with
// all-LDS b64 fragment loads, ping-pong activations) + sigmoid head.
// grid=1, block=256 (8 waves); each wave does 4 of the 32 tiles per layer.
// ---------------------------------------------------------------------------
__global__ __launch_bounds__(256) void mlp_tail_kernel(
    const float* __restrict__ h0,
    const float* __restrict__ w1, const float* __restrict__ bb1,
    const float* __restrict__ w2, const float* __restrict__ bb2,
    const float* __restrict__ w3, const float* __restrict__ bb3,
    const float* __restrict__ w4, const float* __restrict__ bb4,
    const float* __restrict__ w5, const float* __restrict__ bb5,
    float* __restrict__ out)         // (256, 1)
{
    __shared__ float hA[BATCH * 32];
    __shared__ float hB[BATCH * 32];
    __shared__ float Wt[32 * 32];    // current layer weights, transposed [n][k]

    const int tid  = threadIdx.x;
    const int lane = tid & 31;
    const int wv   = tid >> 5;
    const int half = lane >> 4;
    const int lr   = lane & 15;

    for (int i = tid; i < BATCH * 32; i += 256) hA[i] = h0[i];

    const float* Ws[4] = {w1, w2, w3, w4};
    const float* Bs[4] = {bb1, bb2, bb3, bb4};
    float* cur = hA;
    float* nxt = hB;

#pragma unroll
    for (int l = 0; l < 4; ++l) {
        const float* W  = Ws[l];
        const float* Bv = Bs[l];
        // stage transposed weights (barrier below is also the
        // end-of-previous-layer visibility point)
        for (int i = tid; i < 1024; i += 256) {
            int nn = i >> 5, kk = i & 31;
            Wt[nn * 32 + kk] = W[kk * 32 + nn];
        }
        __syncthreads();

#pragma unroll
        for (int tI = 0; tI < 4; ++tI) {
            int tile = wv + 8 * tI;        // 0..31
            int tm = tile & 15, tn = tile >> 4;
            int rm = tm * 16, cn = tn * 16;
            int m = rm + lr, n = cn + lr;

            const float2* Arow = (const float2*)(cur + m * 32) + half;
            const float2* Brow = (const float2*)(Wt  + n * 32) + half;

            v8f acc = {};
#pragma unroll
            for (int k2 = 0; k2 < 16; k2 += 2) {   // K=32 in 8 wmma steps
                float2 a  = Arow[k2];
                float2 bq = Brow[k2];
                v2f A  = {a.x, a.y};
                v2f Bf = {bq.x, bq.y};
                acc = wmma_f32_k4(A, Bf, acc);
            }
            float bn = Bv[n];
#pragma unroll
            for (int r = 0; r < 8; ++r) {
                int mo = rm + r + half * 8;
                float vv = acc[r] + bn;
                nxt[mo * 32 + n] = vv > 0.0f ? vv : 0.0f;
            }
        }
        __syncthreads();
        float* tmp = cur; cur = nxt; nxt = tmp;
    }

    // head: out = sigmoid(h @ w5 + b5)
    if (tid < BATCH) {
        float z = bb5[0];
#pragma unroll
        for (int j = 0; j < 32; ++j) z += cur[tid * 32 + j] * w5[j];
        out[tid] = 1.0f / (1.0f + expf(-z));
    }
}

// ---------------------------------------------------------------------------
extern "C" void kernel_launch(void* const* d_in, const int* in_sizes, int n_in,
                              void* d_out, int out_size, void* d_ws, size_t ws_size,
                              hipStream_t stream) {
    const float* x      = (const float*)d_in[0];
    const float* conv_w = (const float*)d_in[1];
    const float* conv_b = (const float*)d_in[2];
    const float* w0 = (const float*)d_in[3];  const float* b0 = (const float*)d_in[4];
    const float* w1 = (const float*)d_in[5];  const float* b1 = (const float*)d_in[6];
    const float* w2 = (const float*)d_in[7];  const float* b2 = (const float*)d_in[8];
    const float* w3 = (const float*)d_in[9];  const float* b3 = (const float*)d_in[10];
    const float* w4 = (const float*)d_in[11]; const float* b4 = (const float*)d_in[12];
    const float* w5 = (const float*)d_in[13]; const float* b5 = (const float*)d_in[14];
    float* out = (float*)d_out;

    // workspace layout (floats)
    float* sig = (float*)d_ws;                 // 256*SIGP
    float* w0t = sig + BATCH * SIGP;           // 32*SIGP
    float* h0  = w0t + 32 * SIGP;              // 256*32

    sig5_kernel<<<BATCH, 256, 0, stream>>>(x, conv_w, conv_b, sig);
    pack_w0t_kernel<<<(32 * SIGP + 255) / 256, 256, 0, stream>>>(w0, w0t);
    gemm0_kernel<<<4, 256, 0, stream>>>(sig, w0t, b0, h0);
    mlp_tail_kernel<<<1, 256, 0, stream>>>(h0, w1, b1, w2, b2, w3, b3,
                                           w4, b4, w5, b5, out);
    (void)in_sizes; (void)n_in; (void)out_size; (void)ws_size; (void)conv_b;
}